// PMLP_10368051053156
// MI455X (gfx1250) — compile-verified
//
#include <hip/hip_runtime.h>
#include <hip/hip_bf16.h>

typedef float v2f __attribute__((ext_vector_type(2)));
typedef float v8f __attribute__((ext_vector_type(8)));

#define IN_DIM  128
#define HID_DIM 128
#define OUT_DIM 64

// ---------------------------------------------------------------- degree ----
__global__ void init_deg_kernel(float* __restrict__ deg, int n) {
    int i = blockIdx.x * blockDim.x + threadIdx.x;
    if (i < n) deg[i] = 1.0f;                 // self loop contributes 1
}

__global__ void accum_deg_kernel(const int* __restrict__ col,
                                 float* __restrict__ deg, int E) {
    int e = blockIdx.x * blockDim.x + threadIdx.x;
    if (e < E) atomicAdd(&deg[col[e]], 1.0f);
}

__global__ void deg_to_dinv_kernel(float* __restrict__ deg, int n) {
    int i = blockIdx.x * blockDim.x + threadIdx.x;
    if (i < n) deg[i] = rsqrtf(deg[i]);       // deg >= 1 always (self loop)
}

// ------------------------------------------------------------- WMMA GEMM ----
// H[m][n] = sum_k X[m][k] * W[n][k]   (torch convention W is [out,in])
// One wave computes one 16x16 C tile with V_WMMA_F32_16X16X4_F32 over K=128.
// blockDim.x = 32 * (NOUT/16); wave w handles N-tile w; grid.x = M/16 tiles.
template <int NOUT>
__global__ void gemm_wmma_kernel(const float* __restrict__ X,
                                 const float* __restrict__ W,
                                 float* __restrict__ H, int nNodes) {
    constexpr int K = 128;
    const int wv   = threadIdx.x >> 5;        // N-tile index within block
    const int lane = threadIdx.x & 31;
    const int half = lane >> 4;               // 0: K pair {0,1}; 1: K pair {2,3}
    const int l16  = lane & 15;

    const int m0 = blockIdx.x * 16;
    const int n0 = wv * 16;

    // A: 16x4 fp32, lane l16 holds row M=l16, pairs (k0+2*half, k0+2*half+1)
    int mrow = m0 + l16; if (mrow >= nNodes) mrow = nNodes - 1;   // uniform clamp
    const float* aptr = X + (size_t)mrow * K + 2 * half;
    // B: 4x16 fp32, B[k][n] = W[n][k]; lane l16 holds col N=l16, same K pairs
    const float* bptr = W + (size_t)(n0 + l16) * K + 2 * half;

    v8f c = {0.f, 0.f, 0.f, 0.f, 0.f, 0.f, 0.f, 0.f};
#pragma unroll
    for (int k0 = 0; k0 < K; k0 += 4) {
        v2f a = *(const v2f*)(aptr + k0);
        v2f b = *(const v2f*)(bptr + k0);
        // (neg_a, A, neg_b, B, c_mod, C, reuse_a, reuse_b)
        c = __builtin_amdgcn_wmma_f32_16x16x4_f32(false, a, false, b,
                                                  (short)0, c, false, false);
    }

    // C/D layout: VGPR v holds M=v (lanes 0-15) and M=v+8 (lanes 16-31), N=lane&15
    float* out = H + (size_t)m0 * NOUT + n0;
#pragma unroll
    for (int v = 0; v < 8; ++v) {
        int r = v + 8 * half;
        if (m0 + r < nNodes) out[(size_t)r * NOUT + l16] = c[v];
    }
}

// ------------------------------------------------------------- propagate ----
// One wave per edge: agg[col] += dinv[row]*dinv[col] * h[row]
template <int D>
__global__ void propagate_kernel(const int* __restrict__ row,
                                 const int* __restrict__ col,
                                 const float* __restrict__ dinv,
                                 const float* __restrict__ h,
                                 float* __restrict__ agg, int E) {
    int wave = (int)((blockIdx.x * (size_t)blockDim.x + threadIdx.x) >> 5);
    int lane = threadIdx.x & 31;
    if (wave >= E) return;
    int r = row[wave];
    int c = col[wave];
    float nrm = dinv[r] * dinv[c];
    constexpr int V = D / 32;                 // 4 (D=128) or 2 (D=64)
    const float* src = h   + (size_t)r * D + lane * V;
    float*       dst = agg + (size_t)c * D + lane * V;
    if constexpr (V == 4) {
        float4 v = *(const float4*)src;
        atomicAdd(dst + 0, nrm * v.x);
        atomicAdd(dst + 1, nrm * v.y);
        atomicAdd(dst + 2, nrm * v.z);
        atomicAdd(dst + 3, nrm * v.w);
    } else {
        float2 v = *(const float2*)src;
        atomicAdd(dst + 0, nrm * v.x);
        atomicAdd(dst + 1, nrm * v.y);
    }
}

// -------------------------------------------------------------- finalize ----
// out = (relu?)(agg + dinv^2 * h   (self-loop message)  + bias)
template <int D, bool RELU>
__global__ void finalize_kernel(const float* __restrict__ agg,
                                const float* __restrict__ h,
                                const float* __restrict__ dinv,
                                const float* __restrict__ bias,
                                float* __restrict__ out, int n) {
    size_t idx = blockIdx.x * (size_t)blockDim.x + threadIdx.x;
    if (idx >= (size_t)n * D) return;
    int i = (int)(idx / D);
    int d = (int)(idx % D);
    float di = dinv[i];
    float v  = agg[idx] + di * di * h[idx] + bias[d];
    out[idx] = RELU ? fmaxf(v, 0.0f) : v;
}

// ---------------------------------------------------------------- launch ----
extern "C" void kernel_launch(void* const* d_in, const int* in_sizes, int n_in,
                              void* d_out, int out_size, void* d_ws, size_t ws_size,
                              hipStream_t stream) {
    const float* x  = (const float*)d_in[0];
    const int*   ei = (const int*)d_in[1];
    const float* W0 = (const float*)d_in[2];
    const float* b0 = (const float*)d_in[3];
    const float* W1 = (const float*)d_in[4];
    const float* b1 = (const float*)d_in[5];
    const float* W2 = (const float*)d_in[6];
    const float* b2 = (const float*)d_in[7];

    const int N = in_sizes[0] / IN_DIM;       // 50000
    const int E = in_sizes[1] / 2;            // 800000
    const int* rowIdx = ei;                   // edge_index[0]
    const int* colIdx = ei + E;               // edge_index[1]

    // workspace layout
    char*  ws  = (char*)d_ws;
    size_t off = 0;
    auto   alloc = [&](size_t bytes) {
        void* p = ws + off;
        off += (bytes + 255) & ~(size_t)255;
        return p;
    };
    float* dinv = (float*)alloc((size_t)N * sizeof(float));
    float* hbuf = (float*)alloc((size_t)N * HID_DIM * sizeof(float));
    float* agg  = (float*)alloc((size_t)N * HID_DIM * sizeof(float));
    float* xbuf = (float*)alloc((size_t)N * HID_DIM * sizeof(float));

    const int mtiles   = (N + 15) / 16;       // 3125
    const int edgeBlks = (E + 7) / 8;         // 8 waves (edges) per 256-thr block

    // ---- degree / normalization ----
    init_deg_kernel<<<(N + 255) / 256, 256, 0, stream>>>(dinv, N);
    accum_deg_kernel<<<(E + 255) / 256, 256, 0, stream>>>(colIdx, dinv, E);
    deg_to_dinv_kernel<<<(N + 255) / 256, 256, 0, stream>>>(dinv, N);

    // ---- layer 0: 128 -> 128, relu ----
    gemm_wmma_kernel<128><<<mtiles, 256, 0, stream>>>(x, W0, hbuf, N);
    hipMemsetAsync(agg, 0, (size_t)N * 128 * sizeof(float), stream);
    propagate_kernel<128><<<edgeBlks, 256, 0, stream>>>(rowIdx, colIdx, dinv, hbuf, agg, E);
    finalize_kernel<128, true><<<(int)(((size_t)N * 128 + 255) / 256), 256, 0, stream>>>(
        agg, hbuf, dinv, b0, xbuf, N);

    // ---- layer 1: 128 -> 128, relu ----
    gemm_wmma_kernel<128><<<mtiles, 256, 0, stream>>>(xbuf, W1, hbuf, N);
    hipMemsetAsync(agg, 0, (size_t)N * 128 * sizeof(float), stream);
    propagate_kernel<128><<<edgeBlks, 256, 0, stream>>>(rowIdx, colIdx, dinv, hbuf, agg, E);
    finalize_kernel<128, true><<<(int)(((size_t)N * 128 + 255) / 256), 256, 0, stream>>>(
        agg, hbuf, dinv, b1, xbuf, N);

    // ---- layer 2: 128 -> 64, no relu ----
    gemm_wmma_kernel<64><<<mtiles, 128, 0, stream>>>(xbuf, W2, hbuf, N);
    hipMemsetAsync(agg, 0, (size_t)N * 64 * sizeof(float), stream);
    propagate_kernel<64><<<edgeBlks, 256, 0, stream>>>(rowIdx, colIdx, dinv, hbuf, agg, E);
    finalize_kernel<64, false><<<(int)(((size_t)N * 64 + 255) / 256), 256, 0, stream>>>(
        agg, hbuf, dinv, b2, (float*)d_out, N);
}